// Model_68908455297582
// MI455X (gfx1250) — compile-verified
//
#include <hip/hip_runtime.h>
#include <stdint.h>

// Problem constants (match reference)
#define BATCH   64
#define NEV     16
#define NS      32768
#define CHUNK   1024                 // output positions per block
#define NCHUNK  (NS / CHUNK)         // 32 chunks per batch
#define BLK     256                  // 8 wave32s per block
// LDS: 16 event rows x 1024 floats = 64 KB per block (5 blocks/WGP fit in 320 KB)

// ---- monotonic float<->uint key so unsigned atomicMax == float max (exact, handles negatives)
__device__ __forceinline__ uint32_t fkey(float f) {
    uint32_t u = __float_as_uint(f);
    return (u & 0x80000000u) ? ~u : (u | 0x80000000u);
}
__device__ __forceinline__ float fdecode(uint32_t k) {
    uint32_t u = (k & 0x80000000u) ? (k ^ 0x80000000u) : ~k;
    return __uint_as_float(u);
}

// Pass 1: shifted 16-way sum via async global->LDS staging (CDNA5 ASYNCcnt path),
//         write unnormalized sums, block max -> atomicMax(uint key) per batch.
__global__ __launch_bounds__(BLK)
void scatter_sum_kernel(const float* __restrict__ events,
                        const int*   __restrict__ indices,
                        float*       __restrict__ out,
                        uint32_t*    __restrict__ wsmax) {
    __shared__ float lds[NEV * CHUNK];

    const int tid = threadIdx.x;
    const int b   = blockIdx.x / NCHUNK;
    const int j0  = (blockIdx.x % NCHUNK) * CHUNK;

    int idx[NEV];
#pragma unroll
    for (int e = 0; e < NEV; ++e) idx[e] = indices[b * NEV + e];  // uniform per block

    // Low 32 bits of a generic LDS pointer are the LDS byte offset (aperture truncation).
    const uint32_t ldsBase = (uint32_t)(uintptr_t)(&lds[0]);
    const float* base = events + (size_t)b * NEV * NS;

    // ---- stage: async global -> LDS, up to 16 rows in flight (ASYNCcnt-tracked)
#pragma unroll
    for (int e = 0; e < NEV; ++e) {
        const int ie = idx[e];
        if (ie >= j0 + CHUNK) continue;                 // chunk fully before this event's start
        const float*  row  = base + (size_t)e * NS;
        const uint32_t lrow = ldsBase + (uint32_t)(e * CHUNK * 4);
        if (ie <= j0) {
            // fully valid: one coalesced b128 per thread (lane t -> bytes 16t..16t+15)
            const int src = j0 + 4 * tid - ie;          // >= 0, <= NS-4
            uint64_t ga = (uint64_t)(uintptr_t)(row + src);
            uint32_t la = lrow + 16u * (uint32_t)tid;
            asm volatile("global_load_async_to_lds_b128 %0, %1, off"
                         :: "v"(la), "v"(ga) : "memory");
        } else {
            // boundary chunk (rare): per-element b32 with clamped source; invalid slots
            // hold garbage but are never read (predicated at accumulation).
#pragma unroll
            for (int i = 0; i < 4; ++i) {
                int src = j0 + 4 * tid + i - ie;
                if (src < 0) src = 0;
                uint64_t ga = (uint64_t)(uintptr_t)(row + src);
                uint32_t la = lrow + 4u * (uint32_t)(4 * tid + i);
                asm volatile("global_load_async_to_lds_b32 %0, %1, off"
                             :: "v"(la), "v"(ga) : "memory");
            }
        }
    }

    // Each thread reads only LDS slots written by its own async loads -> per-wave wait,
    // no workgroup barrier needed here.
    asm volatile("s_wait_asynccnt 0" ::: "memory");

    // ---- accumulate 16-way sum from LDS (ds_load_b128 per event)
    float a0 = 0.f, a1 = 0.f, a2 = 0.f, a3 = 0.f;
#pragma unroll
    for (int e = 0; e < NEV; ++e) {
        const int ie = idx[e];
        if (ie >= j0 + CHUNK) continue;
        const float4 v = *(const float4*)&lds[e * CHUNK + 4 * tid];
        if (ie <= j0) {
            a0 += v.x; a1 += v.y; a2 += v.z; a3 += v.w;
        } else {
            const int p = j0 + 4 * tid;
            if (p + 0 >= ie) a0 += v.x;
            if (p + 1 >= ie) a1 += v.y;
            if (p + 2 >= ie) a2 += v.z;
            if (p + 3 >= ie) a3 += v.w;
        }
    }

    // unnormalized sums -> d_out (b128 store)
    *(float4*)&out[(size_t)b * NS + j0 + 4 * tid] = make_float4(a0, a1, a2, a3);

    // ---- block max -> per-batch atomic max (uint-key encoding, deterministic)
    float tm = fmaxf(fmaxf(a0, a1), fmaxf(a2, a3));
    __syncthreads();                 // all LDS reads done before reuse
    lds[tid] = tm;
    __syncthreads();
#pragma unroll
    for (int s = BLK / 2; s > 0; s >>= 1) {
        if (tid < s) lds[tid] = fmaxf(lds[tid], lds[tid + s]);
        __syncthreads();
    }
    if (tid == 0) atomicMax(&wsmax[b], fkey(lds[0]));
}

__global__ void init_max_kernel(uint32_t* wsmax) {
    if (threadIdx.x < BATCH) wsmax[threadIdx.x] = 0u;  // key(anything real) > 0
}

// Pass 2: in-place normalize by per-batch (max + 1e-8)
__global__ __launch_bounds__(BLK)
void normalize_kernel(float* __restrict__ out, const uint32_t* __restrict__ wsmax) {
    const size_t gi = ((size_t)blockIdx.x * BLK + threadIdx.x) * 4;  // 4 elems, same batch
    const int b = (int)(gi / NS);
    const float d = fdecode(wsmax[b]) + 1e-8f;
    float4 v = *(const float4*)&out[gi];
    v.x /= d; v.y /= d; v.z /= d; v.w /= d;
    *(float4*)&out[gi] = v;
}

extern "C" void kernel_launch(void* const* d_in, const int* in_sizes, int n_in,
                              void* d_out, int out_size, void* d_ws, size_t ws_size,
                              hipStream_t stream) {
    (void)in_sizes; (void)n_in; (void)out_size; (void)ws_size;
    const float* events  = (const float*)d_in[0];
    const int*   indices = (const int*)d_in[1];
    float*       out     = (float*)d_out;
    uint32_t*    wsmax   = (uint32_t*)d_ws;      // 64 x u32 = 256 B of scratch

    init_max_kernel<<<1, 64, 0, stream>>>(wsmax);
    scatter_sum_kernel<<<BATCH * NCHUNK, BLK, 0, stream>>>(events, indices, out, wsmax);
    normalize_kernel<<<(BATCH * NS) / (BLK * 4), BLK, 0, stream>>>(out, wsmax);
}